// SafeHeteroGraphSAGE_79551384257011
// MI455X (gfx1250) — compile-verified
//
#include <hip/hip_runtime.h>

// ---------------------------------------------------------------------------
// Hetero GraphSAGE (2 layers, C=128) for MI455X / gfx1250.
// Scatter-sum is HBM/atomic bound (~4 GB traffic); GEMMs (~92 GFLOP) use
// V_WMMA_F32_16X16X4_F32 to keep full fp32 accuracy (GEMM time is negligible
// vs. the scatter, so the f32 WMMA flavor is the right precision choice).
// ---------------------------------------------------------------------------

#define CH   128
#define NA   200000
#define NB   200000
#define NC   100000
#define NEDG 500000

typedef __attribute__((ext_vector_type(2))) float v2f;
typedef __attribute__((ext_vector_type(8))) float v8f;

// ---------------- zero fill (float4 per thread) ----------------------------
__global__ void zero_kernel(float* __restrict__ p, int n4) {
  int i = blockIdx.x * blockDim.x + threadIdx.x;
  if (i < n4) ((float4*)p)[i] = make_float4(0.f, 0.f, 0.f, 0.f);
}

// ---------------- edge scatter: agg[dst] += x[src] -------------------------
// One wave32 per edge; each lane moves 4 channels (float4 gather + 4 f32
// atomics). src/dst loads are wave-uniform (broadcast).
__global__ void __launch_bounds__(256)
scatter_kernel(const float* __restrict__ xsrc, const int* __restrict__ src,
               const int* __restrict__ dst, float* __restrict__ agg, int nE) {
  int e = blockIdx.x * 8 + (threadIdx.x >> 5);
  if (e >= nE) return;
  int lane = threadIdx.x & 31;
  int s = src[e];
  int d = dst[e];
  const float4 v = *(const float4*)(xsrc + (size_t)s * CH + lane * 4);
  float* o = agg + (size_t)d * CH + lane * 4;
  atomicAdd(o + 0, v.x);
  atomicAdd(o + 1, v.y);
  atomicAdd(o + 2, v.z);
  atomicAdd(o + 3, v.w);
}

// ---------------- fused SAGE GEMM ------------------------------------------
// out[m,n] += sum_k A1[m,k]*W1[n,k] + sum_k A2[m,k]*W2[n,k] + bias[n]
// M x 128 x 128, fp32 WMMA 16x16x4. One wave per 16-row M tile; block = 8
// waves = 128 rows. A fragments preloaded once per tile; B fragments stream
// from the 64 KB weight matrices (L2/WGP$ resident, shared by all waves).
__global__ void __launch_bounds__(256)
gemm_fused_kernel(const float* __restrict__ A1, const float* __restrict__ W1,
                  const float* __restrict__ A2, const float* __restrict__ W2,
                  const float* __restrict__ bias, float* __restrict__ out,
                  int M) {
  const int lane = threadIdx.x & 31;
  const int wave = threadIdx.x >> 5;
  const int m0   = (blockIdx.x * 8 + wave) * 16;
  if (m0 >= M) return;                 // wave-uniform: EXEC stays all-ones
  const int half = lane >> 4;          // 0: K+{0,1}, 1: K+{2,3} (A/B layout)
  const int l15  = lane & 15;

  // A fragments for all 32 K-steps of this 16x128 tile (ISA 16x4 f32 layout:
  // lane<16 row M=l15 holds K=4kk+{0,1}; lane>=16 holds K=4kk+{2,3}).
  v2f a1[32], a2[32];
  const float* a1p = A1 + (size_t)(m0 + l15) * CH + 2 * half;
  const float* a2p = A2 + (size_t)(m0 + l15) * CH + 2 * half;
#pragma unroll
  for (int kk = 0; kk < 32; ++kk) {
    a1[kk] = *(const v2f*)(a1p + 4 * kk);
    a2[kk] = *(const v2f*)(a2p + 4 * kk);
  }

  for (int nt = 0; nt < 8; ++nt) {
    const int n0 = nt * 16;
    // C/D layout: lane<16 -> rows m0..m0+7, lane>=16 -> rows m0+8..m0+15,
    // column n0+l15; VGPR r = row offset r.
    float* op = out + (size_t)(m0 + 8 * half) * CH + n0 + l15;
    v8f c;
#pragma unroll
    for (int r = 0; r < 8; ++r) c[r] = op[(size_t)r * CH];

    // B[k,n] = W[n,k]: lane indexes n, half selects K+{0,1}/{2,3}.
    const float* b1p = W1 + (size_t)(n0 + l15) * CH + 2 * half;
    const float* b2p = W2 + (size_t)(n0 + l15) * CH + 2 * half;
#pragma unroll
    for (int kk = 0; kk < 32; ++kk) {
      v2f b1 = *(const v2f*)(b1p + 4 * kk);
      c = __builtin_amdgcn_wmma_f32_16x16x4_f32(false, a1[kk], false, b1,
                                                (short)0, c, false, false);
      v2f b2 = *(const v2f*)(b2p + 4 * kk);
      c = __builtin_amdgcn_wmma_f32_16x16x4_f32(false, a2[kk], false, b2,
                                                (short)0, c, false, false);
    }
    const float bb = bias ? bias[n0 + l15] : 0.f;
#pragma unroll
    for (int r = 0; r < 8; ++r) op[(size_t)r * CH] = c[r] + bb;
  }
}

// ---------------- per-node LayerNorm + ReLU --------------------------------
// One wave32 per node row (4 channels/lane); wave reduction via shfl_xor.
__global__ void __launch_bounds__(256)
ln_relu_kernel(const float* __restrict__ x, const float* __restrict__ w,
               const float* __restrict__ b, float* __restrict__ y, int N) {
  int row = blockIdx.x * 8 + (threadIdx.x >> 5);
  if (row >= N) return;
  int lane = threadIdx.x & 31;
  const float4 v = *(const float4*)(x + (size_t)row * CH + lane * 4);
  float s  = v.x + v.y + v.z + v.w;
  float s2 = v.x * v.x + v.y * v.y + v.z * v.z + v.w * v.w;
#pragma unroll
  for (int m = 16; m >= 1; m >>= 1) {
    s  += __shfl_xor(s, m, 32);
    s2 += __shfl_xor(s2, m, 32);
  }
  const float mean = s * (1.f / CH);
  const float var  = s2 * (1.f / CH) - mean * mean;   // ddof=0, matches jnp.var
  const float rstd = rsqrtf(var + 1e-5f);
  const float4 wv = *(const float4*)(w + lane * 4);
  const float4 bv = *(const float4*)(b + lane * 4);
  float4 o;
  o.x = fmaxf((v.x - mean) * rstd * wv.x + bv.x, 0.f);
  o.y = fmaxf((v.y - mean) * rstd * wv.y + bv.y, 0.f);
  o.z = fmaxf((v.z - mean) * rstd * wv.z + bv.z, 0.f);
  o.w = fmaxf((v.w - mean) * rstd * wv.w + bv.w, 0.f);
  *(float4*)(y + (size_t)row * CH + lane * 4) = o;
}

// ---------------------------------------------------------------------------
extern "C" void kernel_launch(void* const* d_in, const int* in_sizes, int n_in,
                              void* d_out, int out_size, void* d_ws,
                              size_t ws_size, hipStream_t stream) {
  (void)in_sizes; (void)n_in; (void)out_size; (void)ws_size;

  const float* x_in[3] = {(const float*)d_in[0], (const float*)d_in[1],
                          (const float*)d_in[2]};
  const int* srcs[4] = {(const int*)d_in[3], (const int*)d_in[5],
                        (const int*)d_in[7], (const int*)d_in[9]};
  const int* dsts[4] = {(const int*)d_in[4], (const int*)d_in[6],
                        (const int*)d_in[8], (const int*)d_in[10]};
  const float* Wl  = (const float*)d_in[11];
  const float* bl  = (const float*)d_in[12];
  const float* Wr  = (const float*)d_in[13];
  const float* lnw = (const float*)d_in[14];
  const float* lnb = (const float*)d_in[15];

  const int Nn[3]    = {NA, NB, NC};
  const int smeta[4] = {0, 1, 1, 2};   // src node type per edge type
  const int dmeta[4] = {1, 0, 2, 1};   // dst node type per edge type

  // Workspace: agg (max-N rows, reused per edge type) + 3 out accumulators.
  float* agg = (float*)d_ws;
  float* outs[3];
  outs[0] = agg + (size_t)NA * CH;
  outs[1] = outs[0] + (size_t)NA * CH;
  outs[2] = outs[1] + (size_t)NB * CH;

  // Activations for both layers live in d_out (layer-1 acts are fully
  // consumed by layer-2 GEMMs before layer-2 LN overwrites them; all kernels
  // serialize on `stream`).
  float* acts[3];
  acts[0] = (float*)d_out;
  acts[1] = acts[0] + (size_t)NA * CH;
  acts[2] = acts[1] + (size_t)NB * CH;

  const float* xs[3] = {x_in[0], x_in[1], x_in[2]};

  for (int l = 0; l < 2; ++l) {
    // zero the three out accumulators
    for (int t = 0; t < 3; ++t) {
      int n4 = Nn[t] * CH / 4;
      zero_kernel<<<(n4 + 255) / 256, 256, 0, stream>>>(outs[t], n4);
    }
    for (int et = 0; et < 4; ++et) {
      const int s = smeta[et], d = dmeta[et];
      int n4 = Nn[d] * CH / 4;
      zero_kernel<<<(n4 + 255) / 256, 256, 0, stream>>>(agg, n4);
      scatter_kernel<<<NEDG / 8, 256, 0, stream>>>(xs[s], srcs[et], dsts[et],
                                                   agg, NEDG);
      const float* W1 = Wl + ((size_t)l * 4 + et) * CH * CH;
      const float* W2 = Wr + ((size_t)l * 4 + et) * CH * CH;
      const float* bb = bl + ((size_t)l * 4 + et) * CH;
      gemm_fused_kernel<<<(Nn[d] + 127) / 128, 256, 0, stream>>>(
          agg, W1, xs[d], W2, bb, outs[d], Nn[d]);
    }
    for (int t = 0; t < 3; ++t) {
      ln_relu_kernel<<<Nn[t] / 8, 256, 0, stream>>>(
          outs[t], lnw + ((size_t)l * 3 + t) * CH,
          lnb + ((size_t)l * 3 + t) * CH, acts[t], Nn[t]);
    }
    xs[0] = acts[0]; xs[1] = acts[1]; xs[2] = acts[2];
  }
}